// Embedding_43447889166721
// MI455X (gfx1250) — compile-verified
//
#include <hip/hip_runtime.h>
#include <hip/hip_bf16.h>

// Embedding row-gather: out[row, :] = table[x[row], :]
//   x:     16384 int32 token ids        (B=8, C=2048 flattened)
//   table: 50257 x 512 float32          (~103 MB -> fits in MI455X 192MB L2)
//   out:   16384 x 512 float32          (32 MiB, streamed once -> NT stores)
//
// Pure memory-bound gather: ~64 MiB total HBM traffic -> ~2.8 us at 23.3 TB/s.
// WMMA (the literal one-hot matmul) would be ~843 GFLOP of wasted ALU work,
// so the optimized CDNA5 kernel is a wave32 wide-vector copy engine with
// scalar (SGPR) row addressing and non-temporal output stores.

typedef float v4f __attribute__((ext_vector_type(4)));

#define EMBED_F32        512          // floats per row
#define WAVES_PER_BLOCK  8            // 256 threads = 8 wave32 on CDNA5
#define BLOCK_THREADS    256

__global__ __launch_bounds__(BLOCK_THREADS)
void embed_gather_kernel(const int* __restrict__ x,
                         const float* __restrict__ table,
                         float* __restrict__ out,
                         int nrows)
{
    const int lane = threadIdx.x & 31;
    // One wave per output row. row is wave-uniform: force it into an SGPR so
    // the token fetch and both base addresses become scalar (saddr VMEM form).
    const int row = __builtin_amdgcn_readfirstlane(
        (int)(blockIdx.x * WAVES_PER_BLOCK + (threadIdx.x >> 5)));
    if (row >= nrows) return;

    // Wave-uniform token id -> SGPR. Compiler can lower this to s_load_b32.
    const int token = __builtin_amdgcn_readfirstlane(x[row]);

    // Row bases in SGPRs; lane offsets are small immediates + lane*16.
    const v4f* __restrict__ src = (const v4f*)(table + (size_t)token * EMBED_F32);
    v4f*       __restrict__ dst = (v4f*)(out   + (size_t)row   * EMBED_F32);

    // 512 floats = 128 float4 per row = 32 lanes x 4 slots.
    // 4 independent global_load_b128 in flight per lane (MLP), then 4
    // non-temporal global_store_b128: output is write-once, never re-read,
    // so keep it out of L2 and preserve the embedding table's residency.
    v4f a0 = src[lane +  0];
    v4f a1 = src[lane + 32];
    v4f a2 = src[lane + 64];
    v4f a3 = src[lane + 96];

    __builtin_nontemporal_store(a0, dst + lane +  0);
    __builtin_nontemporal_store(a1, dst + lane + 32);
    __builtin_nontemporal_store(a2, dst + lane + 64);
    __builtin_nontemporal_store(a3, dst + lane + 96);
}

extern "C" void kernel_launch(void* const* d_in, const int* in_sizes, int n_in,
                              void* d_out, int out_size, void* d_ws, size_t ws_size,
                              hipStream_t stream)
{
    const int*   x     = (const int*)  d_in[0];   // (8, 2048) int32 token ids
    const float* table = (const float*)d_in[1];   // (50257, 512) float32
    float*       out   = (float*)      d_out;     // (8, 2048, 512) float32

    const int nrows  = in_sizes[0];               // 16384 lookups
    const int blocks = (nrows + WAVES_PER_BLOCK - 1) / WAVES_PER_BLOCK;

    embed_gather_kernel<<<blocks, BLOCK_THREADS, 0, stream>>>(x, table, out, nrows);
}